// OSnet_mb_large_sp_ver2_lf_8117488189733
// MI455X (gfx1250) — compile-verified
//
#include <hip/hip_runtime.h>
#include <stdint.h>

// ---------------- problem constants (from reference) ----------------
#define BATCH     32
#define KIDS      256
#define MAXSP     255
#define CNT_SHIFT 44                      // count in bits [63:44]
#define FRAC_SCALE 16777216.0f            // 2^24 fixed-point for the sum
#define FRAC_INV   (1.0f / 16777216.0f)

#define PPB1 32768                        // pixels per block, reduce pass
#define PPB2 16384                        // pixels per block, scatter pass
#define TILE 1024                         // async-staged tile (pixels)

#if defined(__has_builtin)
#  if __has_builtin(__builtin_amdgcn_global_load_async_to_lds_b128)
#    define HAVE_ASYNC_B128 1
#  endif
#endif

typedef int v4i __attribute__((vector_size(16)));
typedef __attribute__((address_space(1))) v4i* as1_v4i_p;   // global
typedef __attribute__((address_space(3))) v4i* as3_v4i_p;   // LDS

__device__ __forceinline__ unsigned long long pack_px(float v) {
    // count=1 in high bits, score as 2^-24 fixed point in low 44 bits
    return (1ull << CNT_SHIFT) | (unsigned long long)(v * FRAC_SCALE);
}

// ---------------- kernel 0: zero the global accumulators ----------------
__global__ void sp_zero(unsigned long long* __restrict__ gacc) {
    gacc[blockIdx.x * 256 + threadIdx.x] = 0ull;
}

// ---------------- kernel 1: per-block LDS histogram, merge to global ----
__global__ __launch_bounds__(256) void sp_accum(const float* __restrict__ score,
                                                const int*   __restrict__ spix,
                                                unsigned long long* __restrict__ gacc,
                                                int npix) {
    __shared__ unsigned long long hist[KIDS];
    const int tid = threadIdx.x;
    const int b   = blockIdx.y;

    hist[tid] = 0ull;                    // blockDim.x == 256 == KIDS
    __syncthreads();

    const size_t base = (size_t)b * (size_t)npix + (size_t)blockIdx.x * PPB1;
    const float* s = score + base;
    const int*   p = spix  + base;

#ifdef HAVE_ASYNC_B128
    // Double-buffered async global->LDS staging (ASYNCcnt path).
    __shared__ float sbuf[2][TILE];
    __shared__ int   ibuf[2][TILE];
    const int ntiles = PPB1 / TILE;      // 32

    auto issue = [&](int t, int buf) {
        const float* gs = s + t * TILE + tid * 4;
        const int*   gp = p + t * TILE + tid * 4;
        __builtin_amdgcn_global_load_async_to_lds_b128(
            (as1_v4i_p)gs, (as3_v4i_p)&sbuf[buf][tid * 4], 0, 0);
        __builtin_amdgcn_global_load_async_to_lds_b128(
            (as1_v4i_p)gp, (as3_v4i_p)&ibuf[buf][tid * 4], 0, 0);
    };

    issue(0, 0);
    for (int t = 0; t < ntiles; ++t) {
        const int cur = t & 1;
        asm volatile("s_wait_asynccnt 0" ::: "memory");  // own wave's tile landed
        __syncthreads();                                 // whole block's tile landed
        if (t + 1 < ntiles) issue(t + 1, cur ^ 1);       // overlap next fetch
#pragma unroll
        for (int e = 0; e < 4; ++e) {
            const int idx = tid * 4 + e;
            const float v = sbuf[cur][idx];
            const int  id = ibuf[cur][idx] & (KIDS - 1);
            atomicAdd(&hist[id], pack_px(v));            // ds_add_u64
        }
        __syncthreads();                                 // tile consumed
    }
#else
    // Fallback: direct b128 vectorized loads.
    const int iters = PPB1 / (256 * 4);                  // 32
    for (int it = 0; it < iters; ++it) {
        const int idx = (it * 256 + tid) * 4;
        const float4 v  = *(const float4*)(s + idx);
        const int4   id = *(const int4*)(p + idx);
        atomicAdd(&hist[id.x & (KIDS - 1)], pack_px(v.x));
        atomicAdd(&hist[id.y & (KIDS - 1)], pack_px(v.y));
        atomicAdd(&hist[id.z & (KIDS - 1)], pack_px(v.z));
        atomicAdd(&hist[id.w & (KIDS - 1)], pack_px(v.w));
    }
#endif

    __syncthreads();
    atomicAdd(&gacc[b * KIDS + tid], hist[tid]);         // global_atomic_add_u64
}

// ---------------- kernel 2: finalize table in LDS + scatter -------------
__global__ __launch_bounds__(256) void sp_scatter(const float* __restrict__ score,
                                                  const int*   __restrict__ spix,
                                                  const unsigned long long* __restrict__ gacc,
                                                  float* __restrict__ out,
                                                  int npix) {
    __shared__ float tab[KIDS];
    const int tid = threadIdx.x;
    const int b   = blockIdx.y;

    {
        const unsigned long long a = gacc[b * KIDS + tid];   // hot in L2
        const unsigned long long cnt = a >> CNT_SHIFT;
        const float sum  = (float)(a & ((1ull << CNT_SHIFT) - 1)) * FRAC_INV;
        const float mean = cnt ? sum / (float)cnt : 0.0f;    // empty seg -> 0 (matches nan>0.5 == false)
        tab[tid] = (mean > 0.5f) ? 1.0f : 0.0f;
    }
    __syncthreads();

    const size_t base = (size_t)b * (size_t)npix + (size_t)blockIdx.x * PPB2;
    const float* s = score + base;
    const int*   p = spix  + base;
    float*       o = out   + base;

    const int iters = PPB2 / (256 * 4);                      // 16
    for (int it = 0; it < iters; ++it) {
        const int idx = (it * 256 + tid) * 4;
        const int4 id = *(const int4*)(p + idx);
        float4 r;
        // score is only read where id==255 (~0.4% of lanes) -> skips ~88% of score traffic
        r.x = (id.x < MAXSP) ? tab[id.x & (KIDS - 1)] : s[idx + 0];
        r.y = (id.y < MAXSP) ? tab[id.y & (KIDS - 1)] : s[idx + 1];
        r.z = (id.z < MAXSP) ? tab[id.z & (KIDS - 1)] : s[idx + 2];
        r.w = (id.w < MAXSP) ? tab[id.w & (KIDS - 1)] : s[idx + 3];
        *(float4*)(o + idx) = r;
    }
}

// ---------------- host-side launch --------------------------------------
extern "C" void kernel_launch(void* const* d_in, const int* in_sizes, int n_in,
                              void* d_out, int out_size, void* d_ws, size_t ws_size,
                              hipStream_t stream) {
    const float* score = (const float*)d_in[0];
    const int*   spix  = (const int*)d_in[1];
    float*       out   = (float*)d_out;
    unsigned long long* gacc = (unsigned long long*)d_ws;    // 32*256*8 = 64 KB

    const int total = in_sizes[0];            // 32 * 1 * 1024 * 1024
    const int npix  = total / BATCH;          // 1,048,576

    sp_zero<<<dim3((BATCH * KIDS) / 256), dim3(256), 0, stream>>>(gacc);

    sp_accum<<<dim3(npix / PPB1, BATCH), dim3(256), 0, stream>>>(score, spix, gacc, npix);

    sp_scatter<<<dim3(npix / PPB2, BATCH), dim3(256), 0, stream>>>(score, spix, gacc, out, npix);
}